// LongDilatedAttentionHead_15839839388312
// MI455X (gfx1250) — compile-verified
//
#include <hip/hip_runtime.h>

typedef __attribute__((ext_vector_type(2))) float v2f;
typedef __attribute__((ext_vector_type(8))) float v8f;
typedef __attribute__((ext_vector_type(4))) unsigned int u32x4;
typedef __attribute__((ext_vector_type(8))) int i32x8;
typedef __attribute__((ext_vector_type(4))) int i32x4;

#define S_LEN   4096
#define E_DIM   1024
#define HD_DIM  64
#define NROWS   (4 * 4096)

#define WMMA_F32(a, b, c) \
    __builtin_amdgcn_wmma_f32_16x16x4_f32(false, (a), false, (b), (short)0, (c), false, false)

// ---------------------------------------------------------------------------
// TDM: build a 2D-tile D# and issue TENSOR_LOAD_TO_LDS (clang-23 6-arg form).
// tensor/tile dims in elements (data_size = 4 bytes), stride in elements,
// lds_off in bytes. Optional LDS row padding (pad codes per ISA 8.4).
// ---------------------------------------------------------------------------
__device__ static inline void tdm_load_2d(unsigned lds_off, const void* gaddr,
                                          unsigned d0, unsigned d1,
                                          unsigned long long stride0,
                                          unsigned pad_en, unsigned pad_interval,
                                          unsigned pad_amount)
{
    unsigned long long ga = (unsigned long long)gaddr;
    u32x4 g0;
    g0[0] = 1u;                                    // count=1, user descriptor
    g0[1] = lds_off;                               // lds_addr (bytes)
    g0[2] = (unsigned)(ga & 0xFFFFFFFFu);          // global_addr[31:0]
    g0[3] = (unsigned)((ga >> 32) & 0x1FFFFFFu)    // global_addr[56:32]
          | (2u << 30);                            // type = 2 ("image")
    i32x8 g1;
    g1[0] = (int)((2u << 16)                       // data_size = 4 bytes
          | (pad_en << 20)
          | (pad_interval << 22)
          | (pad_amount << 25));
    g1[1] = (int)((d0 & 0xFFFFu) << 16);           // tensor_dim0[15:0]
    g1[2] = (int)(((d0 >> 16) & 0xFFFFu) | ((d1 & 0xFFFFu) << 16));
    g1[3] = (int)(((d1 >> 16) & 0xFFFFu) | ((d0 & 0xFFFFu) << 16)); // tile_dim0 = d0
    g1[4] = (int)(d1 & 0xFFFFu);                   // tile_dim1 = d1, tile_dim2 = 0
    g1[5] = (int)(unsigned)(stride0 & 0xFFFFFFFFull);
    g1[6] = (int)(unsigned)((stride0 >> 32) & 0xFFFFull);
    g1[7] = 0;
    i32x4 z4 = (i32x4){0, 0, 0, 0};
    i32x8 z8 = (i32x8){0, 0, 0, 0, 0, 0, 0, 0};
    __builtin_amdgcn_tensor_load_to_lds(g0, g1, z4, z4, z8, 0);
}

// ---------------------------------------------------------------------------
// Kernel 1: fused QKV projection, TDM double-buffered.
// Block: 32 rows x 192 cols (Q|K|V) = 24 16x16 tiles, 3 per wave.
// x chunk [32][64] (+pad -> [32][68]) and W chunks [3][64][64] DMA'd by TDM.
// ---------------------------------------------------------------------------
__global__ __launch_bounds__(256) void qkv_proj_kernel(
    const float* __restrict__ x,
    const float* __restrict__ Wq, const float* __restrict__ bq,
    const float* __restrict__ Wk, const float* __restrict__ bk,
    const float* __restrict__ Wv, const float* __restrict__ bv,
    float* __restrict__ Qo, float* __restrict__ Ko, float* __restrict__ Vo)
{
    __shared__ float xs[2][32][68];      // pad 68: strided col reads hit distinct banks
    __shared__ float wsd[2][3][64][64];

    const int tid  = threadIdx.x;
    const int wave = tid >> 5;
    const int lane = tid & 31;
    const int m16  = lane & 15;
    const int half = lane >> 4;
    const int r0   = blockIdx.x * 32;

    const unsigned xs_off  = (unsigned)(unsigned long long)(void*)&xs[0][0][0];
    const unsigned xs_sz   = 32u * 68u * 4u;
    const unsigned wsd_off = (unsigned)(unsigned long long)(void*)&wsd[0][0][0][0];
    const unsigned w_sz    = 64u * 64u * 4u;

    // tile assignment: 24 tiles (2 row-tiles x 12 col-tiles), 3 per wave
    int rt[3], mat[3], n0[3];
    v8f acc[3];
    #pragma unroll
    for (int i = 0; i < 3; ++i) {
        int t  = wave * 3 + i;
        rt[i]  = t / 12;
        int ct = t % 12;
        mat[i] = ct >> 2;            // 0=Q 1=K 2=V
        n0[i]  = (ct & 3) * 16;
        acc[i] = (v8f){0.f, 0.f, 0.f, 0.f, 0.f, 0.f, 0.f, 0.f};
    }

    // issue chunk 0 into buffer 0 (4 TDM ops per chunk: x + 3x W)
    if (tid == 0) {
        tdm_load_2d(xs_off, x + (size_t)r0 * E_DIM, 64, 32, E_DIM,
                    /*pad*/1, /*interval: 64 dw*/5, /*amount: 4 dw*/3);
        tdm_load_2d(wsd_off + 0 * w_sz, Wq, 64, 64, 64, 0, 0, 0);
        tdm_load_2d(wsd_off + 1 * w_sz, Wk, 64, 64, 64, 0, 0, 0);
        tdm_load_2d(wsd_off + 2 * w_sz, Wv, 64, 64, 64, 0, 0, 0);
    }

    int cur = 0;
    for (int it = 0; it < E_DIM / 64; ++it, cur ^= 1) {
        const int kk0n = (it + 1) * 64;
        if (tid == 0) {
            if (kk0n < E_DIM) {
                // prefetch next chunk into the other buffer, then wait for
                // the current chunk (<=4 outstanding = only the new ops)
                const unsigned nb = (unsigned)(cur ^ 1);
                tdm_load_2d(xs_off + nb * xs_sz,
                            x + (size_t)r0 * E_DIM + kk0n, 64, 32, E_DIM, 1, 5, 3);
                tdm_load_2d(wsd_off + nb * 3u * w_sz + 0 * w_sz,
                            Wq + (size_t)kk0n * HD_DIM, 64, 64, 64, 0, 0, 0);
                tdm_load_2d(wsd_off + nb * 3u * w_sz + 1 * w_sz,
                            Wk + (size_t)kk0n * HD_DIM, 64, 64, 64, 0, 0, 0);
                tdm_load_2d(wsd_off + nb * 3u * w_sz + 2 * w_sz,
                            Wv + (size_t)kk0n * HD_DIM, 64, 64, 64, 0, 0, 0);
                __builtin_amdgcn_s_wait_tensorcnt((short)4);
            } else {
                __builtin_amdgcn_s_wait_tensorcnt((short)0);
            }
        }
        __syncthreads();   // chunk `cur` resident in LDS for all waves

        #pragma unroll
        for (int i = 0; i < 3; ++i) {
            #pragma unroll
            for (int k = 0; k < 64; k += 4) {
                v2f a, b;
                // A frag (16x4 f32): lane = M=m16, K = j + 2*half
                a.x = xs[cur][rt[i] * 16 + m16][k + 2 * half];
                a.y = xs[cur][rt[i] * 16 + m16][k + 2 * half + 1];
                // B frag (4x16): row K = j + 2*half, col N = m16
                b.x = wsd[cur][mat[i]][k + 2 * half][n0[i] + m16];
                b.y = wsd[cur][mat[i]][k + 2 * half + 1][n0[i] + m16];
                acc[i] = WMMA_F32(a, b, acc[i]);
            }
        }
        __syncthreads();   // buffer may be overwritten by next TDM issue
    }

    #pragma unroll
    for (int i = 0; i < 3; ++i) {
        const float* bp = (mat[i] == 0) ? bq : ((mat[i] == 1) ? bk : bv);
        float*       op = (mat[i] == 0) ? Qo : ((mat[i] == 1) ? Ko : Vo);
        float badd = bp[n0[i] + m16];
        #pragma unroll
        for (int v = 0; v < 8; ++v) {   // D: M = v + 8*half, N = m16
            int row = r0 + rt[i] * 16 + v + 8 * half;
            op[(size_t)row * HD_DIM + n0[i] + m16] = acc[i][v] + badd;
        }
    }
}

// ---------------------------------------------------------------------------
// Kernel 2: dilated window attention. Block = 16 query rows; key span = 272
// contiguous positions (17 tiles). Dense WMMA scores + parity mask, LDS
// softmax, dense WMMA attn@V. K/V staged via LDS in 2 chunks (<64 KB total).
// ---------------------------------------------------------------------------
__global__ __launch_bounds__(256) void attn_kernel(
    const float* __restrict__ Qd, const float* __restrict__ Kd,
    const float* __restrict__ Vd, float* __restrict__ out)
{
    __shared__ float KV[144][68];       // pad 68 -> stride 4 banks
    __shared__ float Sc[16][276];       // pad 276 -> stride 20 banks (distinct x16)
    __shared__ float red[2][16][16];

    const int tid  = threadIdx.x;
    const int wave = tid >> 5;
    const int lane = tid & 31;
    const int m16  = lane & 15;
    const int half = lane >> 4;
    const int r0   = blockIdx.x * 16;
    const int bb   = r0 / S_LEN;
    const int s0   = r0 % S_LEN;
    const size_t bbase = (size_t)bb * S_LEN * HD_DIM;

    // Q fragments in registers: A[m16][4*kk + j + 2*half]
    float qx[16], qy[16];
    #pragma unroll
    for (int kk = 0; kk < 16; ++kk) {
        const float* qp = Qd + (size_t)(r0 + m16) * HD_DIM + kk * 4 + 2 * half;
        qx[kk] = qp[0];
        qy[kk] = qp[1];
    }

    // ---- scores: Q[16x64] x K^T[64x272] in 2 key chunks ----
    for (int c = 0; c < 2; ++c) {
        const int start  = c ? 144 : 0;
        const int nk     = c ? 128 : 144;
        const int tb     = c ? 9 : 0;
        const int ntiles = c ? 8 : 9;

        for (int j = tid; j < nk * 16; j += 256) {
            int key = j >> 4, f = j & 15;
            int kt = s0 - 128 + start + key;
            kt = kt < 0 ? 0 : (kt >= S_LEN ? S_LEN - 1 : kt);
            const float* kp = Kd + bbase + (size_t)kt * HD_DIM + f * 4;
            const float4 v4 = *(const float4*)kp;
            KV[key][f * 4 + 0] = v4.x; KV[key][f * 4 + 1] = v4.y;
            KV[key][f * 4 + 2] = v4.z; KV[key][f * 4 + 3] = v4.w;
            // warm L2/L0 for the V pass (global_prefetch_b8)
            __builtin_prefetch(Vd + bbase + (size_t)kt * HD_DIM + f * 4, 0, 1);
        }
        __syncthreads();

        for (int tile = wave; tile < ntiles; tile += 8) {
            int t0g = (tb + tile) * 16;
            int t0l = t0g - start;
            v8f cacc = (v8f){0.f, 0.f, 0.f, 0.f, 0.f, 0.f, 0.f, 0.f};
            #pragma unroll
            for (int kk = 0; kk < 16; ++kk) {
                v2f a, b;
                a.x = qx[kk]; a.y = qy[kk];
                // B[k][n] = K[key(t0+n)][k]
                b.x = KV[t0l + m16][kk * 4 + 2 * half];
                b.y = KV[t0l + m16][kk * 4 + 2 * half + 1];
                cacc = WMMA_F32(a, b, cacc);
            }
            #pragma unroll
            for (int v = 0; v < 8; ++v) {
                int i  = v + 8 * half;          // query row in tile
                int tg = t0g + m16;             // key-local coordinate
                int dt = tg - i;
                int kt = s0 + tg - 128;
                bool valid = (dt >= 0) && (dt <= 256) && ((dt & 1) == 0) &&
                             (kt >= 0) && (kt < S_LEN);
                Sc[i][tg] = valid ? cacc[v] * 0.03125f : -__builtin_inff();
            }
        }
        __syncthreads();
    }

    // ---- softmax over 272 cols, 16 threads per row ----
    {
        int r = tid >> 4, c0 = tid & 15;
        float mx = -__builtin_inff();
        for (int jj = c0; jj < 272; jj += 16) mx = fmaxf(mx, Sc[r][jj]);
        red[0][r][c0] = mx;
        __syncthreads();
        float M = -__builtin_inff();
        #pragma unroll
        for (int k = 0; k < 16; ++k) M = fmaxf(M, red[0][r][k]);
        float sm = 0.f;
        for (int jj = c0; jj < 272; jj += 16) {
            float e = __expf(Sc[r][jj] - M);
            Sc[r][jj] = e;
            sm += e;
        }
        red[1][r][c0] = sm;
        __syncthreads();
        float T = 0.f;
        #pragma unroll
        for (int k = 0; k < 16; ++k) T += red[1][r][k];
        float inv = 1.0f / T;
        for (int jj = c0; jj < 272; jj += 16) Sc[r][jj] *= inv;
    }
    __syncthreads();

    // ---- output: attn[16x272] x V[272x64]; waves 0-3 own 16-col tiles ----
    v8f oacc = (v8f){0.f, 0.f, 0.f, 0.f, 0.f, 0.f, 0.f, 0.f};
    const int h0 = (wave & 3) * 16;
    for (int c = 0; c < 2; ++c) {
        const int start = c ? 144 : 0;
        const int nk    = c ? 128 : 144;
        for (int j = tid; j < nk * 16; j += 256) {
            int key = j >> 4, f = j & 15;
            int kt = s0 - 128 + start + key;
            kt = kt < 0 ? 0 : (kt >= S_LEN ? S_LEN - 1 : kt);
            const float4 v4 = *(const float4*)(Vd + bbase + (size_t)kt * HD_DIM + f * 4);
            KV[key][f * 4 + 0] = v4.x; KV[key][f * 4 + 1] = v4.y;
            KV[key][f * 4 + 2] = v4.z; KV[key][f * 4 + 3] = v4.w;
        }
        __syncthreads();
        if (wave < 4) {
            for (int t = 0; t < nk; t += 4) {
                v2f a, b;
                a.x = Sc[m16][start + t + 2 * half];
                a.y = Sc[m16][start + t + 2 * half + 1];
                b.x = KV[t + 2 * half][h0 + m16];
                b.y = KV[t + 2 * half + 1][h0 + m16];
                oacc = WMMA_F32(a, b, oacc);
            }
        }
        __syncthreads();
    }
    if (wave < 4) {
        #pragma unroll
        for (int v = 0; v < 8; ++v) {
            int row = r0 + v + 8 * half;
            out[(size_t)row * HD_DIM + h0 + m16] = oacc[v];
        }
    }
}

// ---------------------------------------------------------------------------
extern "C" void kernel_launch(void* const* d_in, const int* in_sizes, int n_in,
                              void* d_out, int out_size, void* d_ws, size_t ws_size,
                              hipStream_t stream) {
    const float* x  = (const float*)d_in[0];
    const float* Wq = (const float*)d_in[1];
    const float* bq = (const float*)d_in[2];
    const float* Wk = (const float*)d_in[3];
    const float* bk = (const float*)d_in[4];
    const float* Wv = (const float*)d_in[5];
    const float* bv = (const float*)d_in[6];

    float* Qd = (float*)d_ws;                       // 4 MB
    float* Kd = Qd + (size_t)NROWS * HD_DIM;        // 4 MB
    float* Vd = Kd + (size_t)NROWS * HD_DIM;        // 4 MB
    float* out = (float*)d_out;

    qkv_proj_kernel<<<NROWS / 32, 256, 0, stream>>>(x, Wq, bq, Wk, bk, Wv, bv,
                                                    Qd, Kd, Vd);
    attn_kernel<<<NROWS / 16, 256, 0, stream>>>(Qd, Kd, Vd, out);
}